// GaussSpatialConv_66322884985174
// MI455X (gfx1250) — compile-verified
//
#include <hip/hip_runtime.h>

// ---------------------------------------------------------------------------
// Normalized Gaussian kernel conv (flash-attention style), MI455X / gfx1250.
//
//   out[i,:] = softmax_j( -||x_i - y_j||^2 / (2 s^2) ) @ y_fea
//
// The row-constant -|x_i|^2 term cancels inside the softmax, so the logits
// reduce (in log2 domain) to:   t[i,j] = (x_i . y_j) * C1  -  |y_j|^2 * K2
// with C1 = log2(e)/s^2, K2 = log2(e)/(2 s^2).
//
// Per 16x16 tile:
//   S^T = WMMA_F32_16X16X4_F32( A = [y | y^2*K2] (16j x 4),
//                               B = [x*C1 | -1 ] (4 x 16i) )      -> logits
//   online softmax over j (in-lane over 8 VGPRs + one xor-16 shuffle)
//   O^T += WMMA_F32_16X16X4_F32( A = V^T (16d x 4j), B = P^T (4j x 16i) ) x4
//
// All data is fp32: sigma = 0.1 makes the logits 50x-sensitive to distance
// errors, so fp16 inputs would corrupt exp arguments. fp32 WMMA is cheap
// here because the kernel is transcendental(exp)-bound, not matmul-bound.
// ---------------------------------------------------------------------------

typedef __attribute__((ext_vector_type(2))) float v2f;
typedef __attribute__((ext_vector_type(8))) float v8f;

#define SIGMA   0.1f
#define LOG2E   1.4426950408889634f
#define JCHUNK  128   // j's staged in LDS per block iteration
#define WAVES   4     // waves per block (wave32)

__global__ __launch_bounds__(32 * WAVES)
void gauss_conv_kernel(const float* __restrict__ x,
                       const float* __restrict__ y,
                       const float* __restrict__ yfea,
                       float* __restrict__ out,
                       int N, int M)
{
    const float inv2s2 = 1.0f / (2.0f * SIGMA * SIGMA);   // 50
    const float C1 = 2.0f * inv2s2 * LOG2E;               // log2e / s^2
    const float K2 = inv2s2 * LOG2E;

    // LDS staging: y padded to 4 floats (4th slot = |y|^2 * K2), y_fea row-major.
    __shared__ float sy[JCHUNK * 4];     //  2 KB
    __shared__ float sv[JCHUNK * 16];    //  8 KB

    const int tid  = threadIdx.x;
    const int lane = tid & 31;
    const int wave = tid >> 5;
    const int col  = lane & 15;          // column index inside 16-wide tiles
    const bool hi  = lane >= 16;

    // Each wave owns a 16-row i-tile.
    const int irow = blockIdx.x * (16 * WAVES) + wave * 16 + col;

    // B operand of the logits WMMA: X^T [4k x 16i], constant for whole kernel.
    // Layout (f32 4x16 B): VGPR0 = row k0 (lanes 0-15) / k2 (lanes 16-31),
    //                      VGPR1 = row k1 / k3.  Row k3 = -1 (subtracts y^2*K2).
    v2f bx;
    bx.x = (hi ? x[irow * 3 + 2] : x[irow * 3 + 0]) * C1;
    bx.y = hi ? -1.0f : (x[irow * 3 + 1] * C1);

    v8f acc = {};                 // O^T tile: rows d = v + 8*hi, cols i = col
    float m    = -1e30f;          // running row max (log2 domain), per column i
    float lsum = 0.0f;            // running softmax denominator, per column i

    for (int jb = 0; jb < M; jb += JCHUNK) {
        __syncthreads();
        {   // cooperative stage: one j per thread (M % JCHUNK == 0)
            const int j = jb + tid;
            const float y0 = y[j * 3 + 0];
            const float y1 = y[j * 3 + 1];
            const float y2 = y[j * 3 + 2];
            sy[tid * 4 + 0] = y0;
            sy[tid * 4 + 1] = y1;
            sy[tid * 4 + 2] = y2;
            sy[tid * 4 + 3] = (y0 * y0 + y1 * y1 + y2 * y2) * K2;
            const float4* vsrc = (const float4*)(yfea + j * 16);
            float4* vdst = (float4*)(sv + tid * 16);
            vdst[0] = vsrc[0]; vdst[1] = vsrc[1];
            vdst[2] = vsrc[2]; vdst[3] = vsrc[3];
        }
        __syncthreads();

        for (int jt = 0; jt < JCHUNK / 16; ++jt) {
            const int jr = jt * 16;

            // A operand: Y tile [16j x 4k].  f32 16x4 A layout:
            // VGPR0 = k0 (lanes 0-15) / k2 (lanes 16-31); VGPR1 = k1 / k3.
            v2f ay;
            ay.x = sy[(jr + col) * 4 + (hi ? 2 : 0)];
            ay.y = sy[(jr + col) * 4 + (hi ? 3 : 1)];

            // t = S^T tile of log2-domain logits: lane holds col i = lane%16,
            // rows j = jr + v + 8*hi across the 8 accumulator VGPRs.
            v8f zero = {};
            v8f t = __builtin_amdgcn_wmma_f32_16x16x4_f32(
                        false, ay, false, bx, (short)0, zero, false, false);

            // --- online softmax over j (rows) ---
            float tmax = t[0];
            #pragma unroll
            for (int v = 1; v < 8; ++v) tmax = fmaxf(tmax, t[v]);
            tmax = fmaxf(tmax, __shfl_xor(tmax, 16, 32));   // join lane halves
            const float mnew  = fmaxf(m, tmax);
            const float scale = __builtin_amdgcn_exp2f(m - mnew);
            m = mnew;

            float p[8];
            float psum = 0.0f;
            #pragma unroll
            for (int v = 0; v < 8; ++v) {
                p[v] = __builtin_amdgcn_exp2f(t[v] - mnew);
                psum += p[v];
            }
            psum += __shfl_xor(psum, 16, 32);
            lsum = lsum * scale + psum;
            acc  = acc * scale;                 // per-lane: all cols == i

            // shuffled copies of P^T rows (swap lane halves)
            float ps[8];
            #pragma unroll
            for (int v = 0; v < 8; ++v) ps[v] = __shfl_xor(p[v], 16, 32);

            // --- PV accumulation: O^T += V^T[16d x 4j] * P^T[4j x 16i] ---
            auto pv = [&](int jrow, float b0, float b1, v8f c) -> v8f {
                v2f a, b;
                a.x = sv[(jrow + (hi ? 2 : 0)) * 16 + col];
                a.y = sv[(jrow + (hi ? 3 : 1)) * 16 + col];
                b.x = b0;
                b.y = b1;
                return __builtin_amdgcn_wmma_f32_16x16x4_f32(
                           false, a, false, b, (short)0, c, false, false);
            };
            // P^T rows live as: p[v] -> row (v + 8*hi).  B layout needs
            // VGPR0 = row (4g) lo / (4g+2) hi ; VGPR1 = (4g+1) lo / (4g+3) hi.
            acc = pv(jr + 0,  hi ? ps[2] : p[0],  hi ? ps[3] : p[1],  acc);
            acc = pv(jr + 4,  hi ? ps[6] : p[4],  hi ? ps[7] : p[5],  acc);
            acc = pv(jr + 8,  hi ? p[2]  : ps[0], hi ? p[3]  : ps[1], acc);
            acc = pv(jr + 12, hi ? p[6]  : ps[4], hi ? p[7]  : ps[5], acc);
        }
    }

    // Normalize and store O[i, d]: lane writes d = v + 8*hi for its column i.
    const float inv = 1.0f / lsum;
    float* op = out + irow * 16 + (hi ? 8 : 0);
    float4 o0 = { acc[0] * inv, acc[1] * inv, acc[2] * inv, acc[3] * inv };
    float4 o1 = { acc[4] * inv, acc[5] * inv, acc[6] * inv, acc[7] * inv };
    ((float4*)op)[0] = o0;
    ((float4*)op)[1] = o1;
}

extern "C" void kernel_launch(void* const* d_in, const int* in_sizes, int n_in,
                              void* d_out, int out_size, void* d_ws, size_t ws_size,
                              hipStream_t stream) {
    const float* x    = (const float*)d_in[0];   // [1, N, 3]
    const float* y    = (const float*)d_in[1];   // [1, M, 3]
    const float* yfea = (const float*)d_in[2];   // [1, M, 16]
    float* out = (float*)d_out;                  // [1, N, 16]

    const int N = in_sizes[0] / 3;               // 12288
    const int M = in_sizes[1] / 3;               // 12288

    const int rows_per_block = 16 * WAVES;       // 64
    dim3 grid((N + rows_per_block - 1) / rows_per_block);   // 192 blocks
    dim3 block(32 * WAVES);                      // 128 threads = 4 waves
    gauss_conv_kernel<<<grid, block, 0, stream>>>(x, y, yfea, out, N, M);
}